// EquiNN_25555055411368
// MI455X (gfx1250) — compile-verified
//
#include <hip/hip_runtime.h>
#include <stdint.h>

typedef __attribute__((ext_vector_type(2))) float v2f;
typedef __attribute__((ext_vector_type(4))) float v4f;
typedef __attribute__((ext_vector_type(8))) float v8f;

#define COLS            2048
#define ROWS_PER_BLOCK  16
#define THREADS         256
#define WAVES           8
#define COLS_PER_WAVE   (COLS / WAVES)        // 256
#define LDS_STRIDE      (COLS + 4)            // pad 4 floats -> conflict-free row access
#define TILE_FLOATS     (ROWS_PER_BLOCK * LDS_STRIDE)
#define TILE_VEC4       (ROWS_PER_BLOCK * COLS / 4)   // 8192 b128 chunks
#define SLICE_VEC4      (ROWS_PER_BLOCK * COLS_PER_WAVE / 4)  // 1024 per wave
#define VEC4_PER_SLICE_ROW (COLS_PER_WAVE / 4)        // 64

// out[r][c] = l*X[r][c] + g*rowsum(X[r]) for a (16384 x 2048) fp32 tensor.
// Per 16-row tile (one 256-thread block):
//   Phase 0: each WAVE async-DMAs its own 16x256 slice HBM -> LDS via
//            global_load_async_to_lds_b128; s_wait_asynccnt 0 is per-wave,
//            so no block barrier is needed before that wave starts WMMA.
//   Phase 1: row sums via V_WMMA_F32_16X16X4_F32 with B = ones; one
//            ds_load_b128 feeds TWO wmmas ((f0,f1),(f2,f3)) -- K-slot to
//            column mapping is sum-invariant under B = ones.
//   Phase 2: elementwise from LDS, coalesced global_store_b128.
// Global traffic = exactly 1 read + 1 write per element (roofline minimum:
// ~268 MB / 23.3 TB/s ~= 11.5 us).
__global__ __launch_bounds__(THREADS)
void equinn_fused(const float* __restrict__ X,
                  const float* __restrict__ lp,
                  const float* __restrict__ gp,
                  float* __restrict__ out) {
  extern __shared__ float smem[];
  float* tile    = smem;                                  // [16][LDS_STRIDE]
  float* partial = smem + TILE_FLOATS;                    // [WAVES][16]
  float* rowsum  = partial + WAVES * ROWS_PER_BLOCK;      // [16]

  const int t     = threadIdx.x;
  const int wave  = t >> 5;
  const int lane  = t & 31;
  const int row16 = lane & 15;
  const int hi    = lane >> 4;
  const long r0   = (long)blockIdx.x * ROWS_PER_BLOCK;
  const float* Xt = X + r0 * (long)COLS;
  const int cw    = wave * COLS_PER_WAVE;   // this wave's column base

  // ---- Phase 0: per-wave async DMA of its own 16x256 slice into LDS ----
  {
    const unsigned ldsBase = (unsigned)(uintptr_t)tile;   // low 32 bits = LDS offset
    #pragma unroll 8
    for (int j = 0; j < SLICE_VEC4 / 32; ++j) {           // 32 issues per lane
      const int chunk = j * 32 + lane;                    // [0, 1024)
      const int r = chunk >> 6;                           // 64 vec4 per slice row
      const int c = cw + ((chunk & 63) << 2);             // float column
      const unsigned lds = ldsBase + (unsigned)(r * LDS_STRIDE + c) * 4u;
      const unsigned long long ga =
          (unsigned long long)(uintptr_t)(Xt + (long)r * COLS + c);
      asm volatile("global_load_async_to_lds_b128 %0, %1, off"
                   :: "v"(lds), "v"(ga) : "memory");
    }
    // Per-wave: releases this wave as soon as ITS slice has landed in LDS.
    asm volatile("s_wait_asynccnt 0x0" ::: "memory");
  }

  // ---- Phase 1: WMMA row sums over this wave's own slice ----
  v8f c8 = {};                          // 16x16 f32 accumulator (8 VGPRs)
  v2f b;  b[0] = 1.0f; b[1] = 1.0f;     // B = ones => D[i][j] = sum_k A[i][k]
  const float* aRow = tile + row16 * LDS_STRIDE + cw + 4 * hi;

  #pragma unroll 4
  for (int k = 0; k < COLS_PER_WAVE; k += 8) {
    v4f a4 = *(const v4f*)(aRow + k);   // ds_load_b128, 16B aligned
    v2f a0; a0[0] = a4[0]; a0[1] = a4[1];
    v2f a1; a1[0] = a4[2]; a1[1] = a4[3];
    c8 = __builtin_amdgcn_wmma_f32_16x16x4_f32(false, a0, false, b,
                                               (short)0, c8, false, false);
    c8 = __builtin_amdgcn_wmma_f32_16x16x4_f32(false, a1, false, b,
                                               (short)0, c8, false, false);
  }

  // C/D layout: lanes 0-15 VGPR p -> row p; lanes 16-31 VGPR p -> row p+8.
  if (lane == 0) {
    #pragma unroll
    for (int p = 0; p < 8; ++p) partial[wave * ROWS_PER_BLOCK + p] = c8[p];
  } else if (lane == 16) {
    #pragma unroll
    for (int p = 0; p < 8; ++p) partial[wave * ROWS_PER_BLOCK + 8 + p] = c8[p];
  }
  // Barrier: publishes partials AND makes every wave's async-loaded slice
  // visible block-wide (each wave passed s_wait_asynccnt 0 before arriving).
  __syncthreads();

  if (t < ROWS_PER_BLOCK) {
    float s = 0.0f;
    #pragma unroll
    for (int w = 0; w < WAVES; ++w) s += partial[w * ROWS_PER_BLOCK + t];
    rowsum[t] = s;
  }
  __syncthreads();

  // ---- Phase 2: elementwise from LDS, coalesced b128 stores ----
  const float l = lp[0];
  const float g = gp[0];
  float4* __restrict__ Ov = (float4*)(out + r0 * (long)COLS);

  #pragma unroll 4
  for (int i = t; i < TILE_VEC4; i += THREADS) {
    const int r = i >> 9;               // 512 vec4 per row
    const int c = (i & 511) << 2;
    v4f x = *(const v4f*)(tile + r * LDS_STRIDE + c);   // ds_load_b128
    const float gs = g * rowsum[r];
    float4 o;
    o.x = l * x[0] + gs;
    o.y = l * x[1] + gs;
    o.z = l * x[2] + gs;
    o.w = l * x[3] + gs;
    Ov[i] = o;                                          // global_store_b128
  }
}

extern "C" void kernel_launch(void* const* d_in, const int* in_sizes, int n_in,
                              void* d_out, int out_size, void* d_ws, size_t ws_size,
                              hipStream_t stream) {
  const float* X = (const float*)d_in[0];
  const float* l = (const float*)d_in[1];
  const float* g = (const float*)d_in[2];
  float* out = (float*)d_out;

  const long total  = (long)in_sizes[0];             // 8 * 2048 * 2048
  const int  rows   = (int)(total / COLS);           // 16384
  const int  blocks = rows / ROWS_PER_BLOCK;         // 1024

  const size_t shmem =
      (size_t)(TILE_FLOATS + WAVES * ROWS_PER_BLOCK + ROWS_PER_BLOCK) * sizeof(float);

  equinn_fused<<<blocks, THREADS, shmem, stream>>>(X, l, g, out);
}